// ConvergedInhibition_6030134083807
// MI455X (gfx1250) — compile-verified
//
#include <hip/hip_runtime.h>
#include <hip/hip_bf16.h>

typedef __attribute__((ext_vector_type(2))) float v2f;
typedef __attribute__((ext_vector_type(4))) float v4f;
typedef __attribute__((ext_vector_type(8))) float v8f;

#define SCOPE 63
#define KPAD 64
#define TILE_ROWS 128
#define LDA 68   // padded LDS stride: bank = (4*r + k) % 64 -> conflict-free fragment reads
#define LDG 68

// ---------------------------------------------------------------------------
// Setup: g = real(ifft(1 / fft(delta - filter))), then build padded transposed
// circulant GT[n][k] = g[(n-k) mod 63] (64x64, zero padded). Double-precision
// direct DFT (63^2 terms, trivial cost), on-stream so graph capture works.
// ---------------------------------------------------------------------------
__global__ void semlc_setup_kernel(const float* __restrict__ filt,
                                   float* __restrict__ GT) {
    __shared__ double FFr[SCOPE], FFi[SCOPE];
    __shared__ float  gsh[SCOPE];
    const int t = threadIdx.x;
    const double w0 = 2.0 * 3.14159265358979323846 / (double)SCOPE;

    if (t < SCOPE) {
        // FF[m] = 1 - sum_k f[k] * e^{-i w0 k m}
        double re = 1.0, im = 0.0;
        for (int k = 0; k < SCOPE; ++k) {
            double ang = w0 * (double)k * (double)t;
            double fk  = (double)filt[k];
            re -= fk * cos(ang);
            im += fk * sin(ang);
        }
        FFr[t] = re;
        FFi[t] = im;
    }
    __syncthreads();
    if (t < SCOPE) {
        // g[n] = (1/63) * sum_m Re( e^{+i w0 m n} / FF[m] )
        double acc = 0.0;
        for (int m = 0; m < SCOPE; ++m) {
            double ang = w0 * (double)m * (double)t;
            double den = FFr[m] * FFr[m] + FFi[m] * FFi[m];
            acc += (FFr[m] * cos(ang) + FFi[m] * sin(ang)) / den;
        }
        gsh[t] = (float)(acc / (double)SCOPE);
    }
    __syncthreads();
    for (int i = t; i < KPAD * KPAD; i += blockDim.x) {
        int n = i >> 6, k = i & 63;
        float v = 0.0f;
        if (n < SCOPE && k < SCOPE) v = gsh[(n - k + SCOPE) % SCOPE];
        GT[i] = v;   // GT[n*64 + k] = G[k][n]
    }
}

// ---------------------------------------------------------------------------
// Main GEMM: out[r, 0:63] = A[r, 0:63] x G, via v_wmma_f32_16x16x4_f32.
// Block = 256 threads = 8 waves, 128 rows; wave w owns rows w*16..w*16+15,
// all 64 (padded) output columns in 4 v8f accumulators.
// ---------------------------------------------------------------------------
__global__ __launch_bounds__(256) void semlc_wmma_kernel(
        const float* __restrict__ A, const float* __restrict__ GT,
        float* __restrict__ out, int nrows) {
    __shared__ float sA[TILE_ROWS * LDA];   // ~34 KB
    __shared__ float sG[KPAD * LDG];        // ~17 KB

    const int tid  = threadIdx.x;
    const int lane = tid & 31;
    const int wv   = tid >> 5;
    const int half = lane >> 4;   // 0: lanes 0-15, 1: lanes 16-31
    const int l15  = lane & 15;
    const long tileBase = (long)blockIdx.x * TILE_ROWS;

    // stage GT into padded LDS
    for (int i = tid; i < KPAD * KPAD; i += 256)
        sG[(i >> 6) * LDG + (i & 63)] = GT[i];
    // zero the K=63 pad column of the A tile
    for (int r = tid; r < TILE_ROWS; r += 256)
        sA[r * LDA + SCOPE] = 0.0f;

    // stage A tile: 128 rows x 63 floats = one contiguous 32256 B chunk
    const int rowsHere = (int)((nrows - tileBase) < TILE_ROWS ? (nrows - tileBase)
                                                              : (long)TILE_ROWS);
    const float* src = A + tileBase * SCOPE;
    if (rowsHere == TILE_ROWS) {
        // 2016 aligned float4 loads (32256 % 16 == 0), fully coalesced
        for (int j = tid; j < (TILE_ROWS * SCOPE) / 4; j += 256) {
            v4f x = ((const v4f*)src)[j];
            int i0 = j * 4;
#pragma unroll
            for (int e = 0; e < 4; ++e) {
                int i = i0 + e;
                sA[(i / SCOPE) * LDA + (i % SCOPE)] = x[e];
            }
        }
    } else {
        int nf = rowsHere * SCOPE;
        for (int i = tid; i < nf; i += 256)
            sA[(i / SCOPE) * LDA + (i % SCOPE)] = src[i];
    }
    __syncthreads();

    // A fragment (16x4 f32): lane = M (0-15), v0/v1 = K,K+1; lanes16-31 K+2,K+3
    // B fragment (4x16 f32): lanes = N,  v0/v1 = rows K,K+1 (K+2,K+3 upper half)
    const float* aRow  = &sA[(wv * 16 + l15) * LDA + half * 2];
    const float* gCol0 = &sG[(l15 +  0) * LDG + half * 2];
    const float* gCol1 = &sG[(l15 + 16) * LDG + half * 2];
    const float* gCol2 = &sG[(l15 + 32) * LDG + half * 2];
    const float* gCol3 = &sG[(l15 + 48) * LDG + half * 2];

    v8f c0 = {}, c1 = {}, c2 = {}, c3 = {};
#pragma unroll
    for (int k0 = 0; k0 < KPAD; k0 += 4) {
        v2f a  = *(const v2f*)(aRow  + k0);
        v2f b0 = *(const v2f*)(gCol0 + k0);
        v2f b1 = *(const v2f*)(gCol1 + k0);
        v2f b2 = *(const v2f*)(gCol2 + k0);
        v2f b3 = *(const v2f*)(gCol3 + k0);
        c0 = __builtin_amdgcn_wmma_f32_16x16x4_f32(false, a, false, b0, (short)0, c0, false, false);
        c1 = __builtin_amdgcn_wmma_f32_16x16x4_f32(false, a, false, b1, (short)0, c1, false, false);
        c2 = __builtin_amdgcn_wmma_f32_16x16x4_f32(false, a, false, b2, (short)0, c2, false, false);
        c3 = __builtin_amdgcn_wmma_f32_16x16x4_f32(false, a, false, b3, (short)0, c3, false, false);
    }

    // C layout: VGPR v -> rows v (lanes 0-15) and v+8 (lanes 16-31), col = lane%16
    const long rBase = tileBase + wv * 16 + half * 8;
#pragma unroll
    for (int v = 0; v < 8; ++v) {
        long row = rBase + v;
        if (row < nrows) {
            float* orow = out + row * SCOPE;
            orow[l15]      = c0[v];
            orow[l15 + 16] = c1[v];
            orow[l15 + 32] = c2[v];
            if (l15 + 48 < SCOPE) orow[l15 + 48] = c3[v];  // skip pad col 63
        }
    }
}

extern "C" void kernel_launch(void* const* d_in, const int* in_sizes, int n_in,
                              void* d_out, int out_size, void* d_ws, size_t ws_size,
                              hipStream_t stream) {
    const float* activations = (const float*)d_in[0];
    const float* filt        = (const float*)d_in[1];
    float* out = (float*)d_out;
    float* GT  = (float*)d_ws;               // 64*64 floats = 16 KB scratch

    const int nrows = in_sizes[0] / SCOPE;   // 64*64*128 = 524288

    semlc_setup_kernel<<<1, 64, 0, stream>>>(filt, GT);

    const int blocks = (nrows + TILE_ROWS - 1) / TILE_ROWS;  // 4096
    semlc_wmma_kernel<<<blocks, 256, 0, stream>>>(activations, GT, out, nrows);
}